// MambaBlock_60997125538222
// MI455X (gfx1250) — compile-verified
//
#include <hip/hip_runtime.h>
#include <math.h>
#include <stdint.h>

// ---------------- dimensions ----------------
#define D_MODEL 768
#define D_INNER 1536
#define D_CONV  4
#define DT_RANK 48
#define D_STATE 16
#define BATCH   2
#define SEQLEN  1024
#define BL      (BATCH * SEQLEN)          // 2048 token rows
#define XP_W    (DT_RANK + 2 * D_STATE)   // 80

typedef float v2f __attribute__((ext_vector_type(2)));
typedef float v8f __attribute__((ext_vector_type(8)));

// D = A(16x4,f32) * B(4x16,f32) + C(16x16,f32) -> v_wmma_f32_16x16x4_f32
__device__ __forceinline__ v8f wmma4(v2f a, v2f b, v8f c) {
    return __builtin_amdgcn_wmma_f32_16x16x4_f32(
        false, a, false, b, (short)0, c, false, false);
}

__device__ __forceinline__ float silu_f(float x)     { return x / (1.0f + expf(-x)); }
__device__ __forceinline__ float softplus_f(float x) { return (x > 20.0f) ? x : log1pf(expf(x)); }

// ==================================================================
// Big-GEMM path: block tile 64M x 128N, K-chunks of 32, double-buffered
// LDS staging via CDNA5 global_load_async_to_lds_b128 (ASYNCcnt).
// 8 waves/block: wave_m in 0..3 (16-row strips), wave_n in 0..1 (64 cols,
// 4 WMMA tiles sharing the A fragment).
//   C[M x N] = A[M x K] * W[N x K]^T
// EPI 0: plain store. EPI 1: col<split -> raw C, else SiLU -> C2.
// ==================================================================
#define BM   64
#define BN   128
#define KC   32
#define STR  (KC + 4)    // 36-float LDS row stride: conflict-free fragment reads

template <int EPI>
__global__ __launch_bounds__(256)
void gemm_lds_async(const float* __restrict__ A, const float* __restrict__ W,
                    float* __restrict__ C, float* __restrict__ C2,
                    int K, int lda, int ldw, int ldc, int split) {
    __shared__ float Asm[2][BM][STR];
    __shared__ float Bsm[2][BN][STR];

    const int tid    = threadIdx.x;
    const int lane   = tid & 31;
    const int wave   = tid >> 5;
    const int wave_m = wave & 3;          // 0..3
    const int wave_n = wave >> 2;         // 0..1
    const int row0   = blockIdx.x * BM;   // global M tile origin
    const int col0   = blockIdx.y * BN;   // global N tile origin

    const int m  = lane & 15;             // M idx (A) / N idx (B) within 16-tile
    const int kb = (lane >> 4) << 1;      // per-half K sub-offset (0 or 2)

    // -------- async prefetch of one K-chunk into LDS buffer `buf` --------
    auto issue = [&](int buf, int kc0) {
        const float* gA0 = A + (size_t)row0 * lda + kc0;
        const float* gB0 = W + (size_t)col0 * ldw + kc0;
#pragma unroll
        for (int i = 0; i < 2; ++i) {                     // A: 64 rows * 8 quads
            int slot = tid + i * 256;
            int r = slot >> 3, c4 = (slot & 7) << 2;
            uint32_t l = (uint32_t)(uintptr_t)&Asm[buf][r][c4];
            const float* g = gA0 + (size_t)r * lda + c4;
            asm volatile("global_load_async_to_lds_b128 %0, %1, off"
                         :: "v"(l), "v"(g) : "memory");
        }
#pragma unroll
        for (int i = 0; i < 4; ++i) {                     // B: 128 rows * 8 quads
            int slot = tid + i * 256;
            int r = slot >> 3, c4 = (slot & 7) << 2;
            uint32_t l = (uint32_t)(uintptr_t)&Bsm[buf][r][c4];
            const float* g = gB0 + (size_t)r * ldw + c4;
            asm volatile("global_load_async_to_lds_b128 %0, %1, off"
                         :: "v"(l), "v"(g) : "memory");
        }
    };  // 6 async ops per thread per chunk

    v8f acc[4];
#pragma unroll
    for (int j = 0; j < 4; ++j) acc[j] = (v8f){0.f,0.f,0.f,0.f,0.f,0.f,0.f,0.f};

    const int nchunks = K / KC;
    int buf = 0;
    issue(0, 0);

    for (int kc = 0; kc < nchunks; ++kc) {
        const bool has_next = (kc + 1) < nchunks;
        if (has_next) issue(buf ^ 1, (kc + 1) * KC);
        // in-order per-wave ASYNCcnt retirement: <=6 means current chunk landed
        if (has_next) asm volatile("s_wait_asynccnt 0x6" ::: "memory");
        else          asm volatile("s_wait_asynccnt 0x0" ::: "memory");
        __syncthreads();                               // all waves' copies visible

        const float* As = &Asm[buf][0][0];
        const float* Bs = &Bsm[buf][0][0];
        const float* ar = As + (wave_m * 16 + m) * STR + kb;
#pragma unroll
        for (int kk = 0; kk < KC; kk += 4) {
            v2f a = *(const v2f*)(ar + kk);
#pragma unroll
            for (int j = 0; j < 4; ++j) {
                const float* br = Bs + (wave_n * 64 + j * 16 + m) * STR + kb;
                v2f b = *(const v2f*)(br + kk);
                acc[j] = wmma4(a, b, acc[j]);
            }
        }
        __syncthreads();                               // buffer reusable next iter
        buf ^= 1;
    }

    // C/D layout: VGPR r -> M = r + 8*(lane>=16), N = lane&15
    const int rbase = (lane >> 4) << 3;
#pragma unroll
    for (int j = 0; j < 4; ++j) {
        const int col = col0 + wave_n * 64 + j * 16 + m;
#pragma unroll
        for (int r = 0; r < 8; ++r) {
            const int row = row0 + wave_m * 16 + rbase + r;
            float v = acc[j][r];
            if (EPI == 0) {
                C[(size_t)row * ldc + col] = v;
            } else {
                if (col < split) C [(size_t)row * ldc + col]           = v;
                else             C2[(size_t)row * ldc + (col - split)] = silu_f(v);
            }
        }
    }
}

// ==================================================================
// Small-GEMM path (register-direct WMMA) for xp (N=80) and dt (K=48).
// EPI 0: plain. EPI 2: softplus(v + bias[col]).
// ==================================================================
template <int NB, int EPI>
__global__ void gemm_wmma_f32(const float* __restrict__ A,
                              const float* __restrict__ W,
                              float* __restrict__ C,
                              const float* __restrict__ bias,
                              int M, int N, int K,
                              int lda, int ldw, int ldc) {
    const int lane  = threadIdx.x & 31;
    const int wid   = (blockIdx.x * blockDim.x + threadIdx.x) >> 5;
    const int tilesM  = M >> 4;
    const int ngroups = N / (16 * NB);
    if (wid >= tilesM * ngroups) return;          // wave-uniform

    const int row0 = (wid % tilesM) << 4;
    const int col0 = (wid / tilesM) * 16 * NB;
    const int m  = lane & 15;
    const int kb = (lane >> 4) << 1;

    const float* arow = A + (size_t)(row0 + m) * lda + kb;
    const float* wrow[NB];
#pragma unroll
    for (int j = 0; j < NB; ++j)
        wrow[j] = W + (size_t)(col0 + j * 16 + m) * ldw + kb;

    v8f acc[NB];
#pragma unroll
    for (int j = 0; j < NB; ++j) acc[j] = (v8f){0.f,0.f,0.f,0.f,0.f,0.f,0.f,0.f};

    for (int k = 0; k < K; k += 4) {
        v2f a = *(const v2f*)(arow + k);
#pragma unroll
        for (int j = 0; j < NB; ++j) {
            v2f b = *(const v2f*)(wrow[j] + k);
            acc[j] = wmma4(a, b, acc[j]);
        }
    }

    const int rbase = (lane >> 4) << 3;
#pragma unroll
    for (int j = 0; j < NB; ++j) {
        const int col = col0 + j * 16 + m;
#pragma unroll
        for (int r = 0; r < 8; ++r) {
            const int row = row0 + rbase + r;
            float v = acc[j][r];
            if (EPI == 0) C[(size_t)row * ldc + col] = v;
            else          C[(size_t)row * ldc + col] = softplus_f(v + bias[col]);
        }
    }
}

// ------------------------------------------------------------------
// Depthwise causal conv(4) + SiLU
// ------------------------------------------------------------------
__global__ void conv_silu_kernel(const float* __restrict__ xin,
                                 const float* __restrict__ cw,
                                 const float* __restrict__ cb,
                                 float* __restrict__ xout) {
    const int idx = blockIdx.x * blockDim.x + threadIdx.x;
    if (idx >= BL * D_INNER) return;
    const int d  = idx % D_INNER;
    const int bl = idx / D_INNER;
    const int b  = bl >> 10;
    const int l  = bl & (SEQLEN - 1);

    float acc = cb[d];
#pragma unroll
    for (int j = 0; j < D_CONV; ++j) {
        const int ls = l + j - (D_CONV - 1);
        if (ls >= 0)
            acc = fmaf(cw[d * D_CONV + j],
                       xin[((size_t)(b * SEQLEN + ls)) * D_INNER + d], acc);
    }
    xout[idx] = silu_f(acc);
}

// ------------------------------------------------------------------
// Selective scan: one thread per (b,d); 16-state in registers.
// ------------------------------------------------------------------
__global__ void scan_kernel(const float* __restrict__ dt,
                            const float* __restrict__ xact,
                            const float* __restrict__ resact,
                            const float* __restrict__ xp,
                            const float* __restrict__ A_log,
                            const float* __restrict__ Dvec,
                            float* __restrict__ ybar) {
    const int gid = blockIdx.x * blockDim.x + threadIdx.x;
    if (gid >= BATCH * D_INNER) return;
    const int b = gid / D_INNER;
    const int d = gid % D_INNER;

    float Arow[D_STATE];
#pragma unroll
    for (int n = 0; n < D_STATE; ++n) Arow[n] = -expf(A_log[d * D_STATE + n]);
    const float Dd = Dvec[d];

    float h[D_STATE];
#pragma unroll
    for (int n = 0; n < D_STATE; ++n) h[n] = 0.0f;

    for (int l = 0; l < SEQLEN; ++l) {
        const size_t tok = (size_t)(b * SEQLEN + l);
        const float dtv  = dt  [tok * D_INNER + d];
        const float xv   = xact[tok * D_INNER + d];
        const float* bc  = xp + tok * XP_W + DT_RANK;   // B[0:16], C[16:32]
        const float dbx  = dtv * xv;

        float y = 0.0f;
#pragma unroll
        for (int n = 0; n < D_STATE; ++n) {
            const float dA = __expf(dtv * Arow[n]);
            h[n] = fmaf(dA, h[n], dbx * bc[n]);
            y    = fmaf(h[n], bc[D_STATE + n], y);
        }
        ybar[tok * D_INNER + d] = (y + xv * Dd) * resact[tok * D_INNER + d];
    }
}

// ------------------------------------------------------------------
extern "C" void kernel_launch(void* const* d_in, const int* in_sizes, int n_in,
                              void* d_out, int out_size, void* d_ws, size_t ws_size,
                              hipStream_t stream) {
    const float* hs     = (const float*)d_in[0];  // (2,1024,768)
    const float* W_in   = (const float*)d_in[1];  // (3072,768)
    const float* conv_w = (const float*)d_in[2];  // (1536,1,4)
    const float* conv_b = (const float*)d_in[3];  // (1536)
    const float* W_x    = (const float*)d_in[4];  // (80,1536)
    const float* W_dt   = (const float*)d_in[5];  // (1536,48)
    const float* b_dt   = (const float*)d_in[6];  // (1536)
    const float* A_log  = (const float*)d_in[7];  // (1536,16)
    const float* Dv     = (const float*)d_in[8];  // (1536)
    const float* W_out  = (const float*)d_in[9];  // (768,1536)
    float* out = (float*)d_out;                   // (2,1024,768)

    float* ws      = (float*)d_ws;
    const size_t T = (size_t)BL * D_INNER;
    float* proj_x  = ws;                          // raw x half of in_proj
    float* res_act = proj_x + T;                  // silu(res)
    float* x_act   = res_act + T;                 // conv+silu
    float* xp      = x_act + T;                   // (2048,80)
    float* ybar    = xp + (size_t)BL * XP_W;
    float* dtbuf   = proj_x;                      // alias: proj_x dead after conv

    const int THREADS = 256;

    // 1) in_proj (2048x3072, K=768): async-LDS WMMA, SiLU-split epilogue
    gemm_lds_async<1><<<dim3(BL / BM, (2 * D_INNER) / BN), THREADS, 0, stream>>>(
        hs, W_in, proj_x, res_act, D_MODEL, D_MODEL, D_MODEL, D_INNER, D_INNER);

    // 2) depthwise conv + SiLU
    {
        const int n = BL * D_INNER;
        conv_silu_kernel<<<(n + THREADS - 1) / THREADS, THREADS, 0, stream>>>(
            proj_x, conv_w, conv_b, x_act);
    }
    // 3) xp = x_act @ W_x^T  (2048x80, K=1536)
    {
        const int waves = (BL / 16) * (XP_W / 16);
        gemm_wmma_f32<1, 0><<<(waves + 7) / 8, THREADS, 0, stream>>>(
            x_act, W_x, xp, nullptr, BL, XP_W, D_INNER, D_INNER, D_INNER, XP_W);
    }
    // 4) dt = softplus(xp[:, :48] @ W_dt^T + b_dt)  (2048x1536, K=48)
    {
        const int waves = (BL / 16) * (D_INNER / 64);
        gemm_wmma_f32<4, 2><<<(waves + 7) / 8, THREADS, 0, stream>>>(
            xp, W_dt, dtbuf, b_dt, BL, D_INNER, DT_RANK, XP_W, DT_RANK, D_INNER);
    }
    // 5) selective scan
    {
        const int n = BATCH * D_INNER;
        scan_kernel<<<(n + THREADS - 1) / THREADS, THREADS, 0, stream>>>(
            dtbuf, x_act, res_act, xp, A_log, Dv, ybar);
    }
    // 6) out = ybar @ W_out^T  (2048x768, K=1536): async-LDS WMMA
    gemm_lds_async<0><<<dim3(BL / BM, D_MODEL / BN), THREADS, 0, stream>>>(
        ybar, W_out, out, nullptr, D_INNER, D_INNER, D_INNER, D_MODEL, 0);
}